// AttentionFocalLoss_80625126080788
// MI455X (gfx1250) — compile-verified
//
#include <hip/hip_runtime.h>

// ---------------------------------------------------------------------------
// AttentionFocalLoss for MI455X (gfx1250, wave32).
// Memory-bound streaming reduction: 79.3 MB of f32 preds read once
// (~3.4 us at 23.3 TB/s HBM). Inner loop kept branchless and cheap
// (~12 VALU + 3 TRANS per element) so the kernel stays on the HBM roofline.
// Wave reduction uses V_WMMA_F32_16X16X4_F32 (B = ones) as an exact f32
// cross-lane adder: D[m,n] = partial[m] + partial[m+16].
// ---------------------------------------------------------------------------

#define BATCH 64
#define NGT   32
#define NCH   80          // cate_out_channels
#define SSZ   512.0f
#define SIG   0.2f

// cells per level (B*g*g) cumulative offsets; total = 247808 = 968*256 exactly
#define CO1 102400
#define CO2 185344
#define CO3 222208
#define CO4 238592
#define TOT_CELLS 247808
#define NBLK_CELLS 968    // TOT_CELLS / 256

// prior index level boundaries (from ANCHOR_MARK)
#define PM0 24576
#define PM1 30720
#define PM2 32256
#define PM3 32640
#define NPRI 32736

typedef __attribute__((ext_vector_type(2))) float v2f;
typedef __attribute__((ext_vector_type(8))) float v8f;

// --------------------------- init ------------------------------------------
__global__ void k_init(unsigned* __restrict__ hit, int* __restrict__ num_pos) {
    int i = threadIdx.x;
    if (i < 5 * BATCH) hit[i] = 0u;
    if (i == 5 * BATCH) *num_pos = 0;
}

// --------------------------- hit masks -------------------------------------
// hit[lv*64 + b] bit j  <=>  some prior of level lv in batch b matched gt j
__global__ void k_hit(const int* __restrict__ bti, unsigned* __restrict__ hit) {
    __shared__ unsigned lm[5];
    int b     = blockIdx.x >> 7;     // 128 chunks of 256 priors per batch
    int chunk = blockIdx.x & 127;
    int p     = chunk * 256 + threadIdx.x;
    if (threadIdx.x < 5) lm[threadIdx.x] = 0u;
    __syncthreads();
    if (p < NPRI) {
        int lv = p < PM0 ? 0 : p < PM1 ? 1 : p < PM2 ? 2 : p < PM3 ? 3 : 4;
        int j  = bti[b * NPRI + p];
        if (j >= 0) atomicOr(&lm[lv], 1u << j);
    }
    __syncthreads();
    if (threadIdx.x < 5 && lm[threadIdx.x])
        atomicOr(&hit[threadIdx.x * BATCH + b], lm[threadIdx.x]);
}

// --------------------------- label assignment ------------------------------
__global__ void k_labels(const float* __restrict__ targets,
                         const unsigned* __restrict__ hit,
                         unsigned char* __restrict__ labels,
                         int* __restrict__ num_pos) {
    int idx = blockIdx.x * 256 + threadIdx.x;      // exact, no tail
    int lv, off, g;
    if      (idx < CO1) { lv = 0; off = 0;   g = 40; }
    else if (idx < CO2) { lv = 1; off = CO1; g = 36; }
    else if (idx < CO3) { lv = 2; off = CO2; g = 24; }
    else if (idx < CO4) { lv = 3; off = CO3; g = 16; }
    else                { lv = 4; off = CO4; g = 12; }
    int gg    = g * g;
    int local = idx - off;
    int b     = local / gg;
    int cell  = local - b * gg;
    int y     = cell / g;
    int x     = cell - y * g;

    unsigned hm  = hit[lv * BATCH + b];
    float   invg = 1.0f / (float)g;
    int     best = -1;

    for (int j = 0; j < NGT; ++j) {
        if (!((hm >> j) & 1u)) continue;
        const float* t = targets + (b * NGT + j) * 5;
        float x1 = t[0] * SSZ, y1 = t[1] * SSZ, x2 = t[2] * SSZ, y2 = t[3] * SSZ;
        float hw = 0.5f * (x2 - x1) * SIG;
        float hh = 0.5f * (y2 - y1) * SIG;
        float cw = 0.5f * (x2 + x1);
        float ch = 0.5f * (y2 + y1);
        int cwq   = (int)floorf((cw / SSZ) / invg);
        int chq   = (int)floorf((ch / SSZ) / invg);
        int topb  = max(0,     (int)floorf(((ch - hh) / SSZ) / invg));
        int downb = min(g - 1, (int)floorf(((ch + hh) / SSZ) / invg));
        int leftb = max(0,     (int)floorf(((cw - hw) / SSZ) / invg));
        int rightb= min(g - 1, (int)floorf(((cw + hw) / SSZ) / invg));
        int top   = max(topb,  chq - 1);
        int down  = min(downb, chq + 1);
        int left  = max(leftb, cwq - 1);
        int right = min(rightb, cwq + 1);
        if (y >= top && y <= down && x >= left && x <= right) best = j;
    }
    int lab = (best >= 0) ? (int)targets[(b * NGT + best) * 5 + 4] : 0;
    labels[idx] = (unsigned char)lab;

    unsigned long long bal = __ballot(lab > 0);
    if ((threadIdx.x & 31) == 0)
        atomicAdd(num_pos, (int)__popcll(bal));
}

// --------------------------- focal loss ------------------------------------
__device__ __forceinline__ float wave_sum_wmma(float partial) {
#if __has_builtin(__builtin_amdgcn_wmma_f32_16x16x4_f32)
    // A(16x4): lane L<16 -> A[L,0..1], lane L>=16 -> A[L-16,2..3].
    // With B = all-ones: D[m,n] = partial[m] + partial[m+16]  (exact f32).
    v2f a;  a.x = partial; a.y = 0.0f;
    v2f bo; bo.x = 1.0f;   bo.y = 1.0f;
    v8f c = {};
    v8f d = __builtin_amdgcn_wmma_f32_16x16x4_f32(false, a, false, bo,
                                                  (short)0, c, false, false);
    // lane<16 holds rows 0..7, lane>=16 holds rows 8..15 in its 8 D-VGPRs
    float s8 = d[0] + d[1] + d[2] + d[3] + d[4] + d[5] + d[6] + d[7];
    return s8 + __shfl_xor(s8, 16, 32);
#else
    float s = partial;
    for (int m = 16; m >= 1; m >>= 1) s += __shfl_xor(s, m, 32);
    return s;
#endif
}

__global__ void k_focal(const float* __restrict__ p0, const float* __restrict__ p1,
                        const float* __restrict__ p2, const float* __restrict__ p3,
                        const float* __restrict__ p4,
                        const unsigned char* __restrict__ labels,
                        float* __restrict__ blockSums) {
    int idx = blockIdx.x * 256 + threadIdx.x;      // one thread per grid cell
    const float* pred; int off, gg;
    if      (idx < CO1) { pred = p0; off = 0;   gg = 1600; }
    else if (idx < CO2) { pred = p1; off = CO1; gg = 1296; }
    else if (idx < CO3) { pred = p2; off = CO2; gg = 576;  }
    else if (idx < CO4) { pred = p3; off = CO3; gg = 256;  }
    else                { pred = p4; off = CO4; gg = 144;  }
    int local = idx - off;
    int b     = local / gg;
    int cell  = local - b * gg;
    const float* base = pred + (size_t)b * NCH * gg + cell;
    int lab = (int)labels[idx];

    float acc = 0.0f;
    for (int c = 0; c < NCH; ++c) {
        __builtin_prefetch(base + (size_t)(c + 1) * gg, 0, 0);
        float v   = base[(size_t)c * gg];            // coalesced across the wave
        // shared transcendentals for both class arms (branchless select below)
        float enx = __expf(-fabsf(v));               // e^{-|v|} in (0,1]
        float opx = 1.0f + enx;                      // in (1,2] -> no cancellation
        float l1p = __logf(opx);                     // == log1p(e^{-|v|})
        float inv = __builtin_amdgcn_rcpf(opx);
        float p   = (v >= 0.0f) ? inv : enx * inv;   // sigmoid(v)
        bool  t   = (lab == c + 1);
        float lg  = (t ? fminf(v, 0.0f) : fminf(-v, 0.0f)) - l1p; // log-sigmoid arm
        float q   = t ? (1.0f - p) : p;              // (1-p) or p
        float co  = t ? 0.25f : 0.75f;               // alpha / (1-alpha)
        acc -= co * q * q * lg;                      // pure v_cndmask path, no branch
    }

    // exact f32 wave reduction via WMMA (EXEC all ones: block is full, uniform)
    float wsum = wave_sum_wmma(acc);

    __shared__ float wpart[8];
    int wid = threadIdx.x >> 5;
    if ((threadIdx.x & 31) == 0) wpart[wid] = wsum;
    __syncthreads();
    if (threadIdx.x == 0) {
        float s = 0.0f;
        for (int i = 0; i < 8; ++i) s += wpart[i];
        blockSums[blockIdx.x] = s;                  // deterministic per-block
    }
}

// --------------------------- final reduce + divide -------------------------
__global__ void k_final(const float* __restrict__ blockSums,
                        const int* __restrict__ num_pos,
                        float* __restrict__ out) {
    __shared__ float sm[256];
    float s = 0.0f;
    for (int i = threadIdx.x; i < NBLK_CELLS; i += 256) s += blockSums[i];
    sm[threadIdx.x] = s;
    __syncthreads();
    for (int st = 128; st > 0; st >>= 1) {
        if (threadIdx.x < st) sm[threadIdx.x] += sm[threadIdx.x + st];
        __syncthreads();
    }
    if (threadIdx.x == 0)
        out[0] = sm[0] / (float)(*num_pos + 1);
}

// --------------------------- launcher --------------------------------------
extern "C" void kernel_launch(void* const* d_in, const int* in_sizes, int n_in,
                              void* d_out, int out_size, void* d_ws, size_t ws_size,
                              hipStream_t stream) {
    const float* p0      = (const float*)d_in[0];
    const float* p1      = (const float*)d_in[1];
    const float* p2      = (const float*)d_in[2];
    const float* p3      = (const float*)d_in[3];
    const float* p4      = (const float*)d_in[4];
    const float* targets = (const float*)d_in[5];
    const int*   bti     = (const int*)d_in[6];

    char* ws = (char*)d_ws;
    int*           num_pos   = (int*)(ws + 0);
    unsigned*      hit       = (unsigned*)(ws + 64);        // 5*64 u32 = 1280 B
    float*         blockSums = (float*)(ws + 2048);         // 968 f32  = 3872 B
    unsigned char* labels    = (unsigned char*)(ws + 8192); // 247808 B
    // total workspace use: 256000 bytes

    k_init  <<<1, 5 * BATCH + 32, 0, stream>>>(hit, num_pos);
    k_hit   <<<BATCH * 128, 256, 0, stream>>>(bti, hit);
    k_labels<<<NBLK_CELLS, 256, 0, stream>>>(targets, hit, labels, num_pos);
    k_focal <<<NBLK_CELLS, 256, 0, stream>>>(p0, p1, p2, p3, p4, labels, blockSums);
    k_final <<<1, 256, 0, stream>>>(blockSums, num_pos, (float*)d_out);
}